// GCNModule_47777216200968
// MI455X (gfx1250) — compile-verified
//
#include <hip/hip_runtime.h>
#include <hip/hip_bf16.h>

// GCN 2-layer forward for MI455X (gfx1250), wave32.
//   h   = X @ W                       (f32 WMMA 16x16x4, LDS-staged operands)
//   acc = scatter-add over edges of h[src] * dinv[src]*dinv[dst]
//   out = acc + h*dinv^2 (self loop) + bias  [+ ReLU for layer 1]

typedef __attribute__((ext_vector_type(2))) float v2f;
typedef __attribute__((ext_vector_type(8))) float v8f;
typedef __attribute__((ext_vector_type(4))) int   v4i;

// Typed address-space pointers for the async LDS-copy builtin
// (clang signature: (v4i AS1*, v4i AS3*, imm int offset, imm int cpol)).
typedef __attribute__((address_space(1))) v4i* gp_v4i;
typedef __attribute__((address_space(3))) v4i* lp_v4i;

#define N_NODES 50000
#define N_EDGES 800000
#define DIM 128
#define NH (N_NODES * DIM)   // 6,400,000

#if defined(__has_builtin)
#if __has_builtin(__builtin_amdgcn_global_load_async_to_lds_b128)
#define HAVE_ASYNC_LDS 1
#endif
#endif

// ---------------------------------------------------------------------------
// init: deg <- 1.0 (self loop), acc <- 0, out <- 0
// ---------------------------------------------------------------------------
__global__ void init_kernel(float* __restrict__ deg, float* __restrict__ acc,
                            float* __restrict__ out) {
  int idx = blockIdx.x * blockDim.x + threadIdx.x;
  if (idx < NH) {
    acc[idx] = 0.0f;
    out[idx] = 0.0f;
  }
  if (idx < N_NODES) deg[idx] = 1.0f;
}

// deg[dst] += 1 for every edge
__global__ void degree_kernel(const int* __restrict__ dst, float* __restrict__ deg) {
  int e = blockIdx.x * blockDim.x + threadIdx.x;
  if (e < N_EDGES) atomicAdd(&deg[dst[e]], 1.0f);
}

// in-place deg -> rsqrt(deg)
__global__ void rsqrt_kernel(float* __restrict__ d) {
  int i = blockIdx.x * blockDim.x + threadIdx.x;
  if (i < N_NODES) d[i] = rsqrtf(d[i]);
}

// ---------------------------------------------------------------------------
// GEMM: D[M x 128] = A[M x 128] @ B[128 x 128], all row-major f32.
// Block = 256 threads = 8 waves; block owns one 16-row M stripe, wave w owns
// N-tile w. M = 50000 = 3125*16 exactly -> no divergence, EXEC all 1s (WMMA
// requirement). A stripe (8KB) and all of B (64KB) are staged into LDS once
// per block — via GLOBAL_LOAD_ASYNC_TO_LDS_B128 (ASYNCcnt) when the builtin
// exists, else plain float4 copies.
//
// V_WMMA_F32_16X16X4_F32 fragment layouts (ISA 7.12.2, 32-bit A 16x4):
//   A: lanes 0-15 -> M=lane, VGPR{0,1} = K{k0,k0+1}; lanes 16-31 -> K{k0+2,k0+3}
//   B: (4x16) lanes 0-15 -> N=lane, VGPR{0,1} = K{k0,k0+1}; lanes 16-31 -> +2
//   C/D: VGPR r -> (M = r + 8*(lane>=16), N = lane%16)
// ---------------------------------------------------------------------------
__global__ void gemm_wmma_f32(const float* __restrict__ A, const float* __restrict__ B,
                              float* __restrict__ D) {
  __shared__ float sA[16 * DIM];    //  8 KB: this block's A stripe
  __shared__ float sB[DIM * DIM];   // 64 KB: full weight matrix

  const int t     = threadIdx.x;
  const int lane  = t & 31;
  const int wave  = t >> 5;                  // 0..7 -> N tile
  const int m0    = blockIdx.x * 16;
  const int n0    = wave * 16;
  const int sub   = lane & 15;               // M row (A) / N col (B)
  const int khalf = lane >> 4;               // 0 or 1 -> K pair select

  const float4* gA  = (const float4*)(A + (long)m0 * DIM);  // 512 float4
  const float4* gB  = (const float4*)B;                     // 4096 float4
  float4*       s4A = (float4*)sA;
  float4*       s4B = (float4*)sB;

#if HAVE_ASYNC_LDS
#pragma unroll
  for (int i = t; i < 512; i += 256)
    __builtin_amdgcn_global_load_async_to_lds_b128(
        (gp_v4i)(gA + i), (lp_v4i)(s4A + i), 0, 0);
#pragma unroll
  for (int i = t; i < 4096; i += 256)
    __builtin_amdgcn_global_load_async_to_lds_b128(
        (gp_v4i)(gB + i), (lp_v4i)(s4B + i), 0, 0);
#if __has_builtin(__builtin_amdgcn_s_wait_asynccnt)
  __builtin_amdgcn_s_wait_asynccnt(0);
#else
  asm volatile("s_wait_asynccnt 0x0" ::: "memory");
#endif
#else
#pragma unroll
  for (int i = t; i < 512; i += 256) s4A[i] = gA[i];
#pragma unroll
  for (int i = t; i < 4096; i += 256) s4B[i] = gB[i];
#endif
  __syncthreads();

  const float* arow = sA + sub * DIM;   // A row for this lane (ds_load_b64)
  const float* bcol = sB + n0 + sub;    // B column base       (ds_load_b32 x2)

  v8f c = {};
#pragma unroll
  for (int k0 = 0; k0 < DIM; k0 += 4) {
    const int ka = k0 + khalf * 2;
    v2f a, b;
    a.x = arow[ka];
    a.y = arow[ka + 1];
    b.x = bcol[ka * DIM];
    b.y = bcol[(ka + 1) * DIM];
    c = __builtin_amdgcn_wmma_f32_16x16x4_f32(
        /*neg_a=*/false, a, /*neg_b=*/false, b,
        /*c_mod=*/(short)0, c, /*reuse_a=*/false, /*reuse_b=*/false);
  }

  const int mbase = m0 + khalf * 8;
#pragma unroll
  for (int r = 0; r < 8; ++r) {
    D[(long)(mbase + r) * DIM + n0 + sub] = c[r];
  }
}

// ---------------------------------------------------------------------------
// Edge aggregation: one wave32 per edge; lane l covers features 4l..4l+3.
// acc[dst] += h[src] * dinv[src]*dinv[dst]
// h (25.6MB) and acc (25.6MB) are both L2-resident (192MB L2), so the random
// gather + f32 atomic scatter runs at L2 rather than HBM latitude.
// ---------------------------------------------------------------------------
__global__ void edge_aggregate(const float* __restrict__ h, const int* __restrict__ src,
                               const int* __restrict__ dst, const float* __restrict__ dinv,
                               float* __restrict__ acc) {
  const int e    = (blockIdx.x * blockDim.x + threadIdx.x) >> 5;
  const int lane = threadIdx.x & 31;
  if (e >= N_EDGES) return;
  const int s = src[e];
  const int d = dst[e];
  const float w = dinv[s] * dinv[d];
  const float4 v = ((const float4*)(h + (long)s * DIM))[lane];
  float* o = acc + (long)d * DIM + 4 * lane;
  atomicAdd(o + 0, v.x * w);
  atomicAdd(o + 1, v.y * w);
  atomicAdd(o + 2, v.z * w);
  atomicAdd(o + 3, v.w * w);
}

// ---------------------------------------------------------------------------
// finalize: out = acc + h * dinv^2 (self loop) + bias, optional ReLU.
// Safe in-place (out may alias acc): pure elementwise.
// ---------------------------------------------------------------------------
__global__ void finalize_kernel(const float* __restrict__ acc, const float* __restrict__ h,
                                const float* __restrict__ dinv, const float* __restrict__ bias,
                                float* __restrict__ out, int relu) {
  const int idx = blockIdx.x * blockDim.x + threadIdx.x;
  if (idx >= NH) return;
  const int node = idx >> 7;
  const int f    = idx & (DIM - 1);
  const float di = dinv[node];
  float v = acc[idx] + h[idx] * di * di + bias[f];
  out[idx] = relu ? fmaxf(v, 0.0f) : v;
}

// ---------------------------------------------------------------------------
extern "C" void kernel_launch(void* const* d_in, const int* in_sizes, int n_in,
                              void* d_out, int out_size, void* d_ws, size_t ws_size,
                              hipStream_t stream) {
  (void)in_sizes; (void)n_in; (void)out_size; (void)ws_size;

  const float* x   = (const float*)d_in[0];
  const int*   ei  = (const int*)d_in[1];      // [2, E]: row 0 = src, row 1 = dst
  const float* W1  = (const float*)d_in[2];
  const float* b1  = (const float*)d_in[3];
  const float* W2  = (const float*)d_in[4];
  const float* b2  = (const float*)d_in[5];
  float*       out = (float*)d_out;

  const int* src = ei;
  const int* dst = ei + N_EDGES;

  // Workspace layout (floats): [dinv | hbuf | accbuf]  ~51.5 MB total
  float* ws     = (float*)d_ws;
  float* dinv   = ws;                 // N_NODES (padded to 65536)
  float* hbuf   = ws + 65536;         // NH
  float* accbuf = hbuf + NH;          // NH

  const int T = 256;

  // --- normalization: deg = 1 + indegree(dst); dinv = rsqrt(deg)
  init_kernel<<<(NH + T - 1) / T, T, 0, stream>>>(dinv, accbuf, out);
  degree_kernel<<<(N_EDGES + T - 1) / T, T, 0, stream>>>(dst, dinv);
  rsqrt_kernel<<<(N_NODES + T - 1) / T, T, 0, stream>>>(dinv);

  // --- layer 1: h = x @ W1 ; aggregate ; + self-loop + b1 ; ReLU -> accbuf
  gemm_wmma_f32<<<N_NODES / 16, T, 0, stream>>>(x, W1, hbuf);
  edge_aggregate<<<(N_EDGES * 32) / T, T, 0, stream>>>(hbuf, src, dst, dinv, accbuf);
  finalize_kernel<<<NH / T, T, 0, stream>>>(accbuf, hbuf, dinv, b1, accbuf, 1);

  // --- layer 2: h2 = h1 @ W2 ; aggregate into d_out ; + self-loop + b2
  gemm_wmma_f32<<<N_NODES / 16, T, 0, stream>>>(accbuf, W2, hbuf);
  edge_aggregate<<<(N_EDGES * 32) / T, T, 0, stream>>>(hbuf, src, dst, dinv, out);
  finalize_kernel<<<NH / T, T, 0, stream>>>(out, hbuf, dinv, b2, out, 0);
}